// BoidPolicy_62758062129379
// MI455X (gfx1250) — compile-verified
//
#include <hip/hip_runtime.h>

// Boids all-pairs policy, N = 8192, fp32.
//   Kernel 1: split-K pairwise accumulation, j-tiles staged to LDS via the
//             CDNA5 Tensor Data Mover (tensor_load_to_lds + s_wait_tensorcnt).
//   Kernel 2: reduce partials + normalize/weight/noise/clip epilogue.

#define BN     8192
#define BLK    256
#define NCHUNK 8
#define CHUNK  (BN / NCHUNK)          // 1024 boids per j-chunk

typedef unsigned int u32x4 __attribute__((ext_vector_type(4)));
typedef int          i32x8 __attribute__((ext_vector_type(8)));
typedef int          i32x4 __attribute__((ext_vector_type(4)));

#if defined(__HIP_DEVICE_COMPILE__) && defined(__gfx1250__) && \
    __has_builtin(__builtin_amdgcn_tensor_load_to_lds)
#define USE_TDM 1
#else
#define USE_TDM 0
#endif

#if USE_TDM
// 1-D contiguous DMA of `n` f32 elements from global `gsrc` into LDS `lds_dst`.
// D# per CDNA5 ISA ch.8: group0 = {count=1 | lds_addr | global_addr | type=2},
// group1 = {data_size=4B, tensor_dim0 = tile_dim0 = n, stride0 = n, 1 row}.
// Groups 2/3 (dims >= 3) are disabled -> zero vectors.
__device__ __forceinline__ void tdm_load_1d(void* lds_dst, const void* gsrc,
                                            unsigned n) {
  unsigned long long ga = (unsigned long long)gsrc;
  unsigned lds_off = (unsigned)(unsigned long long)lds_dst;  // low 32 = LDS byte offset
  u32x4 g0;
  g0.x = 1u;                                           // count=1, user descriptor
  g0.y = lds_off;                                      // lds_addr
  g0.z = (unsigned)(ga & 0xFFFFFFFFull);               // global_addr[31:0]
  g0.w = (unsigned)((ga >> 32) & 0x1FFFFFFull) | (2u << 30);  // addr[56:32] | type=2
  i32x8 g1;
  g1[0] = (int)(2u << 16);                   // wg_mask=0, data_size=2 (4 bytes)
  g1[1] = (int)((n & 0xFFFFu) << 16);        // tensor_dim0[15:0]  (bits 63:48)
  g1[2] = (int)((n >> 16) & 0xFFFFu);        // tensor_dim0[31:16]; tensor_dim1=0
  g1[3] = (int)((n & 0xFFFFu) << 16);        // tile_dim0 = n      (bits 127:112)
  g1[4] = 0;                                 // tile_dim1=0, tile_dim2=0 (1-D)
  g1[5] = (int)n;                            // tensor_dim0_stride[31:0]
  g1[6] = 0;                                 // stride0[47:32]=0, stride1 lo=0
  g1[7] = 0;
  i32x4 z4 = {0, 0, 0, 0};
  i32x8 z8 = {0, 0, 0, 0, 0, 0, 0, 0};
  __builtin_amdgcn_tensor_load_to_lds(g0, g1, z4, z4, z8, 0);
}
#endif

__global__ __launch_bounds__(BLK)
void boids_pairs(const float* __restrict__ pos, const float* __restrict__ vel,
                 float* __restrict__ part) {
  __shared__ __align__(16) float lds_p[CHUNK * 2];
  __shared__ __align__(16) float lds_v[CHUNK * 2];
  const int i     = blockIdx.x * BLK + threadIdx.x;   // this thread's boid
  const int jbase = blockIdx.y * CHUNK;               // j-chunk base

#if USE_TDM
  if (threadIdx.x == 0) {   // one DMA per array; TDM ignores EXEC
    tdm_load_1d(lds_p, pos + (size_t)jbase * 2, CHUNK * 2);
    tdm_load_1d(lds_v, vel + (size_t)jbase * 2, CHUNK * 2);
  }
#if __has_builtin(__builtin_amdgcn_s_wait_tensorcnt)
  __builtin_amdgcn_s_wait_tensorcnt(0);
#else
  asm volatile("s_wait_tensorcnt 0x0" ::: "memory");
#endif
#else
  { // fallback: cooperative staged copy
    const float2* gp = (const float2*)(pos + (size_t)jbase * 2);
    const float2* gv = (const float2*)(vel + (size_t)jbase * 2);
    for (int t = threadIdx.x; t < CHUNK; t += BLK) {
      ((float2*)lds_p)[t] = gp[t];
      ((float2*)lds_v)[t] = gv[t];
    }
  }
#endif
  __syncthreads();

  const float pix = pos[(size_t)i * 2 + 0];
  const float piy = pos[(size_t)i * 2 + 1];
  float sx = 0.f, sy = 0.f, cnt = 0.f;
  float vxs = 0.f, vys = 0.f, pxs = 0.f, pys = 0.f;

#pragma unroll 8
  for (int j = 0; j < CHUNK; ++j) {
    float2 pj = ((const float2*)lds_p)[j];   // broadcast LDS read
    float2 vj = ((const float2*)lds_v)[j];
    float dx = pj.x - pix;
    float dy = pj.y - piy;
    dx -= __builtin_rintf(dx);               // toroidal wrap (v_rndne_f32);
    dy -= __builtin_rintf(dy);               // ties-to-even matches strict >0.5 test
    float d2 = __builtin_fmaf(dx, dx, dy * dy);
    float sm = (d2 <= 0.0004f) ? 1.0f : 0.0f;  // SEP_DIST^2
    float pm = (d2 <= 0.04f)   ? 1.0f : 0.0f;  // PERCEPTION^2
    // self-pair (dx=dy=0) contributes 0 to all diff-sums; cnt/vel self terms
    // are removed in the finalize kernel -> no per-pair index compare needed.
    sx  = __builtin_fmaf(sm, -dx, sx);
    sy  = __builtin_fmaf(sm, -dy, sy);
    cnt += pm;
    vxs = __builtin_fmaf(pm, vj.x, vxs);
    vys = __builtin_fmaf(pm, vj.y, vys);
    pxs = __builtin_fmaf(pm, dx, pxs);
    pys = __builtin_fmaf(pm, dy, pys);
  }

  float4* rec = (float4*)(part + ((size_t)blockIdx.y * BN + i) * 8);
  rec[0] = make_float4(sx, sy, cnt, vxs);
  rec[1] = make_float4(vys, pxs, pys, 0.f);
}

__global__ __launch_bounds__(BLK)
void boids_finalize(const float* __restrict__ part, const float* __restrict__ vel,
                    const float* __restrict__ noise,
                    const float* __restrict__ w_sep, const float* __restrict__ w_ali,
                    const float* __restrict__ w_coh, const float* __restrict__ n_scale,
                    float* __restrict__ out) {
  const int i = blockIdx.x * BLK + threadIdx.x;
  float sx = 0.f, sy = 0.f, cnt = 0.f;
  float vxs = 0.f, vys = 0.f, pxs = 0.f, pys = 0.f;
  const float4* p4 = (const float4*)part;
#pragma unroll
  for (int c = 0; c < NCHUNK; ++c) {
    float4 a = p4[((size_t)c * BN + i) * 2 + 0];
    float4 b = p4[((size_t)c * BN + i) * 2 + 1];
    sx += a.x; sy += a.y; cnt += a.z; vxs += a.w;
    vys += b.x; pxs += b.y; pys += b.z;
  }
  const float vix = vel[(size_t)i * 2 + 0];
  const float viy = vel[(size_t)i * 2 + 1];
  cnt -= 1.0f;            // remove self from perception accumulators
  vxs -= vix;
  vys -= viy;

  const float EPSF = 1e-8f;
  // separation steer
  float dsep = fmaxf(sqrtf(sx * sx + sy * sy), EPSF);
  float nsx = sx / dsep, nsy = sy / dsep;
  // alignment steer
  float ic = 1.0f / cnt;
  float ax = vxs * ic - vix, ay = vys * ic - viy;
  float da = fmaxf(sqrtf(ax * ax + ay * ay), EPSF);
  ax /= da; ay /= da;
  // cohesion steer
  float cx = pxs * ic, cy = pys * ic;
  float dc = fmaxf(sqrtf(cx * cx + cy * cy), EPSF);
  cx /= dc; cy /= dc;

  float wS = *w_sep, wA = *w_ali, wC = *w_coh, nS = *n_scale;
  float accx = wS * nsx + wA * ax + wC * cx + nS * noise[(size_t)i * 2 + 0];
  float accy = wS * nsy + wA * ay + wC * cy + nS * noise[(size_t)i * 2 + 1];

  float an = sqrtf(accx * accx + accy * accy);
  if (an > 1.0f) {        // ACC_MAX = 1
    float s = 1.0f / fmaxf(an, EPSF);
    accx *= s; accy *= s;
  }
  out[(size_t)i * 2 + 0] = accx;
  out[(size_t)i * 2 + 1] = accy;
}

extern "C" void kernel_launch(void* const* d_in, const int* in_sizes, int n_in,
                              void* d_out, int out_size, void* d_ws, size_t ws_size,
                              hipStream_t stream) {
  const float* pos   = (const float*)d_in[0];
  const float* vel   = (const float*)d_in[1];
  const float* noise = (const float*)d_in[2];
  const float* wS    = (const float*)d_in[3];
  const float* wA    = (const float*)d_in[4];
  const float* wC    = (const float*)d_in[5];
  const float* nS    = (const float*)d_in[6];
  float* part = (float*)d_ws;   // NCHUNK * BN * 8 floats = 2 MiB, fully overwritten

  dim3 grid1(BN / BLK, NCHUNK);
  boids_pairs<<<grid1, BLK, 0, stream>>>(pos, vel, part);
  boids_finalize<<<BN / BLK, BLK, 0, stream>>>(part, vel, noise, wS, wA, wC, nS,
                                               (float*)d_out);
  (void)in_sizes; (void)n_in; (void)out_size; (void)ws_size;
}